// GCN_79319456023095
// MI455X (gfx1250) — compile-verified
//
#include <hip/hip_runtime.h>
#include <hip/hip_bf16.h>

#define N_NODES 50000
#define N_EDGES 1600000
#define N_OBS   128
#define HIDDEN  128
#define N_ACT   32
#define K_OUT   64

typedef __attribute__((ext_vector_type(2))) float v2f;
typedef __attribute__((ext_vector_type(8))) float v8f;

// ---- workspace layout (in floats) ----
// deg   : [N]            degree accumulator
// dinv  : [N]            1/sqrt(deg+1)
// hW1   : [N,128]        x^T @ W1
// agg1  : [N,128]        edge aggregate -> becomes h1 (relu'd) in place
// hW2   : [N,32]         h1 @ W2
// agg2  : [N,32]         edge aggregate -> becomes h2 in place
static constexpr size_t OFF_DEG  = 0;
static constexpr size_t OFF_DINV = 50176;                  // 50000 padded
static constexpr size_t OFF_HW1  = OFF_DINV + 50176;
static constexpr size_t OFF_AGG1 = OFF_HW1 + (size_t)N_NODES * HIDDEN;
static constexpr size_t OFF_HW2  = OFF_AGG1 + (size_t)N_NODES * HIDDEN;
static constexpr size_t OFF_AGG2 = OFF_HW2 + (size_t)N_NODES * N_ACT;

__global__ void zero_kernel(float* __restrict__ p, int n) {
    int i = blockIdx.x * blockDim.x + threadIdx.x;
    if (i < n) p[i] = 0.0f;
}

__global__ void deg_kernel(const int* __restrict__ dst, float* __restrict__ deg) {
    int e = blockIdx.x * blockDim.x + threadIdx.x;
    if (e < N_EDGES) atomicAdd(&deg[dst[e]], 1.0f);
}

__global__ void dinv_kernel(const float* __restrict__ deg, float* __restrict__ dinv) {
    int n = blockIdx.x * blockDim.x + threadIdx.x;
    if (n < N_NODES) dinv[n] = rsqrtf(deg[n] + 1.0f);   // deg>=0 -> >=1, safe
}

// hW1[n][j] = sum_k x[k][n] * W1[k][j]   (x is [128, N] row-major)
// wave = one 16x16 output tile; 8 waves/block cover all j tiles of a node tile
__global__ __launch_bounds__(256)
void gemm1_kernel(const float* __restrict__ x, const float* __restrict__ W1,
                  float* __restrict__ hW1) {
    const int wave = threadIdx.x >> 5;          // j tile 0..7
    const int lane = threadIdx.x & 31;
    const int n0   = blockIdx.x * 16;           // node tile (50000 = 16*3125)
    const int j0   = wave * 16;
    const int m    = lane & 15;                 // A row / B col within tile
    const int kp   = (lane >> 4) * 2;           // K pair: 0 or 2
    v8f c = {};
    #pragma unroll 4
    for (int k0 = 0; k0 < N_OBS; k0 += 4) {
        v2f a, b;
        a.x = x[(size_t)(k0 + kp)     * N_NODES + n0 + m];
        a.y = x[(size_t)(k0 + kp + 1) * N_NODES + n0 + m];
        b.x = W1[(k0 + kp)     * HIDDEN + j0 + m];
        b.y = W1[(k0 + kp + 1) * HIDDEN + j0 + m];
        c = __builtin_amdgcn_wmma_f32_16x16x4_f32(false, a, false, b,
                                                  (short)0, c, false, false);
    }
    #pragma unroll
    for (int r = 0; r < 8; ++r) {
        const int mm = r + 8 * (lane >> 4);     // C/D layout: VGPR r = rows r, r+8
        hW1[(size_t)(n0 + mm) * HIDDEN + j0 + m] = c[r];
    }
}

// agg1[dst] += hW1[src] * dinv[src]*dinv[dst] ; one wave per edge, float4 per lane
__global__ __launch_bounds__(256)
void edge1_kernel(const int* __restrict__ src, const int* __restrict__ dst,
                  const float* __restrict__ dinv, const float* __restrict__ hW1,
                  float* __restrict__ agg1) {
    const int lane = threadIdx.x & 31;
    const int nw   = (gridDim.x * blockDim.x) >> 5;
    for (int e = (blockIdx.x * blockDim.x + threadIdx.x) >> 5; e < N_EDGES; e += nw) {
        const int s = src[e], d = dst[e];
        const float norm = dinv[s] * dinv[d];
        const float4 v = *(const float4*)(hW1 + (size_t)s * HIDDEN + lane * 4);
        float* ap = agg1 + (size_t)d * HIDDEN + lane * 4;
        atomicAdd(ap + 0, v.x * norm);
        atomicAdd(ap + 1, v.y * norm);
        atomicAdd(ap + 2, v.z * norm);
        atomicAdd(ap + 3, v.w * norm);
    }
}

// h1 = relu(agg1 + hW1*dinv^2 + b1), written back into agg1
__global__ void fin1_kernel(const float* __restrict__ hW1, const float* __restrict__ dinv,
                            const float* __restrict__ b1, float* __restrict__ agg1) {
    int i = blockIdx.x * blockDim.x + threadIdx.x;
    if (i < N_NODES * HIDDEN) {
        const int n = i >> 7, j = i & 127;
        const float di = dinv[n];
        const float v = agg1[i] + hW1[i] * di * di + b1[j];
        agg1[i] = fmaxf(v, 0.0f);
    }
}

// hW2[n][j] = sum_k h1[n][k] * W2[k][j]   (h1 row-major [N,128], W2 [128,32])
__global__ __launch_bounds__(64)
void gemm2_kernel(const float* __restrict__ h1, const float* __restrict__ W2,
                  float* __restrict__ hW2) {
    const int wave = threadIdx.x >> 5;          // j tile 0..1
    const int lane = threadIdx.x & 31;
    const int n0   = blockIdx.x * 16;
    const int j0   = wave * 16;
    const int m    = lane & 15;
    const int kp   = (lane >> 4) * 2;
    v8f c = {};
    #pragma unroll 4
    for (int k0 = 0; k0 < HIDDEN; k0 += 4) {
        const float2 av = *(const float2*)(h1 + (size_t)(n0 + m) * HIDDEN + k0 + kp);
        v2f a, b;
        a.x = av.x; a.y = av.y;
        b.x = W2[(k0 + kp)     * N_ACT + j0 + m];
        b.y = W2[(k0 + kp + 1) * N_ACT + j0 + m];
        c = __builtin_amdgcn_wmma_f32_16x16x4_f32(false, a, false, b,
                                                  (short)0, c, false, false);
    }
    #pragma unroll
    for (int r = 0; r < 8; ++r) {
        const int mm = r + 8 * (lane >> 4);
        hW2[(size_t)(n0 + mm) * N_ACT + j0 + m] = c[r];
    }
}

// agg2[dst] += hW2[src] * norm ; one wave per edge, 1 float per lane
__global__ __launch_bounds__(256)
void edge2_kernel(const int* __restrict__ src, const int* __restrict__ dst,
                  const float* __restrict__ dinv, const float* __restrict__ hW2,
                  float* __restrict__ agg2) {
    const int lane = threadIdx.x & 31;
    const int nw   = (gridDim.x * blockDim.x) >> 5;
    for (int e = (blockIdx.x * blockDim.x + threadIdx.x) >> 5; e < N_EDGES; e += nw) {
        const int s = src[e], d = dst[e];
        const float norm = dinv[s] * dinv[d];
        const float v = hW2[(size_t)s * N_ACT + lane] * norm;
        atomicAdd(agg2 + (size_t)d * N_ACT + lane, v);
    }
}

// h2 = agg2 + hW2*dinv^2 + b2, written back into agg2 (no relu on layer 2)
__global__ void fin2_kernel(const float* __restrict__ hW2, const float* __restrict__ dinv,
                            const float* __restrict__ b2, float* __restrict__ agg2) {
    int i = blockIdx.x * blockDim.x + threadIdx.x;
    if (i < N_NODES * N_ACT) {
        const int n = i >> 5, j = i & 31;
        const float di = dinv[n];
        agg2[i] = agg2[i] + hW2[i] * di * di + b2[j];
    }
}

// out[64,32] = onehot[64,N] @ h2[N,32]; K split into 400-node chunks (125 blocks),
// 8 waves/block = 4 M-tiles x 2 N-tiles; partials reduced via fp32 atomics.
__global__ __launch_bounds__(256)
void gemm_out_kernel(const float* __restrict__ onehot, const float* __restrict__ h2,
                     float* __restrict__ out) {
    const int wave = threadIdx.x >> 5;
    const int lane = threadIdx.x & 31;
    const int m0   = (wave >> 1) * 16;          // K_OUT tile
    const int j0   = (wave & 1) * 16;           // N_ACT tile
    const int m    = lane & 15;
    const int kp   = (lane >> 4) * 2;
    const int nbase = blockIdx.x * 400;         // 125 * 400 = 50000
    v8f c = {};
    for (int n = nbase; n < nbase + 400; n += 4) {
        const float2 av = *(const float2*)(onehot + (size_t)(m0 + m) * N_NODES + n + kp);
        v2f a, b;
        a.x = av.x; a.y = av.y;
        b.x = h2[(size_t)(n + kp)     * N_ACT + j0 + m];
        b.y = h2[(size_t)(n + kp + 1) * N_ACT + j0 + m];
        c = __builtin_amdgcn_wmma_f32_16x16x4_f32(false, a, false, b,
                                                  (short)0, c, false, false);
    }
    #pragma unroll
    for (int r = 0; r < 8; ++r) {
        const int mm = r + 8 * (lane >> 4);
        atomicAdd(out + (m0 + mm) * N_ACT + j0 + m, c[r]);
    }
}

extern "C" void kernel_launch(void* const* d_in, const int* in_sizes, int n_in,
                              void* d_out, int out_size, void* d_ws, size_t ws_size,
                              hipStream_t stream) {
    const float* x      = (const float*)d_in[0];
    const int*   ei     = (const int*)d_in[1];
    const float* onehot = (const float*)d_in[2];
    const float* W1     = (const float*)d_in[3];
    const float* b1     = (const float*)d_in[4];
    const float* W2     = (const float*)d_in[5];
    const float* b2     = (const float*)d_in[6];
    const int* src = ei;
    const int* dst = ei + N_EDGES;
    float* out = (float*)d_out;

    float* ws   = (float*)d_ws;
    float* deg  = ws + OFF_DEG;
    float* dinv = ws + OFF_DINV;
    float* hW1  = ws + OFF_HW1;
    float* agg1 = ws + OFF_AGG1;   // becomes h1
    float* hW2  = ws + OFF_HW2;
    float* agg2 = ws + OFF_AGG2;   // becomes h2

    // --- zero accumulators + output ---
    zero_kernel<<<(N_NODES + 255) / 256, 256, 0, stream>>>(deg, N_NODES);
    zero_kernel<<<(N_NODES * HIDDEN + 255) / 256, 256, 0, stream>>>(agg1, N_NODES * HIDDEN);
    zero_kernel<<<(N_NODES * N_ACT + 255) / 256, 256, 0, stream>>>(agg2, N_NODES * N_ACT);
    zero_kernel<<<(K_OUT * N_ACT + 255) / 256, 256, 0, stream>>>(out, K_OUT * N_ACT);

    // --- degree / normalization ---
    deg_kernel<<<(N_EDGES + 255) / 256, 256, 0, stream>>>(dst, deg);
    dinv_kernel<<<(N_NODES + 255) / 256, 256, 0, stream>>>(deg, dinv);

    // --- layer 1 ---
    gemm1_kernel<<<N_NODES / 16, 256, 0, stream>>>(x, W1, hW1);
    edge1_kernel<<<8192, 256, 0, stream>>>(src, dst, dinv, hW1, agg1);
    fin1_kernel<<<(N_NODES * HIDDEN + 255) / 256, 256, 0, stream>>>(hW1, dinv, b1, agg1);

    // --- layer 2 ---
    gemm2_kernel<<<N_NODES / 16, 64, 0, stream>>>(agg1, W2, hW2);
    edge2_kernel<<<8192, 256, 0, stream>>>(src, dst, dinv, hW2, agg2);
    fin2_kernel<<<(N_NODES * N_ACT + 255) / 256, 256, 0, stream>>>(hW2, dinv, b2, agg2);

    // --- readout ---
    gemm_out_kernel<<<N_NODES / 400, 256, 0, stream>>>(onehot, agg2, out);
}